// PolarizedSelfAttention_30743375905253
// MI455X (gfx1250) — compile-verified
//
#include <hip/hip_runtime.h>

// ---------------------------------------------------------------------------
// Problem constants (from reference): B=8, L=4096, C=1024, H=512.
// Exact-math simplification: output[b,l,c] == sigma[b] (one scalar per batch).
//   sigma[b] = sigmoid( <u@w4+b4 , u@w5+b5> ), u = sigmoid(z@w3+b3),
//   z = y@w2+b2, y = softmax(x@w1+b1 over L)-weighted sum of x over L.
// Bandwidth roofline: 128 MB read + 128 MB re-read (L2-resident; x=128MB<192MB L2)
// + 128 MB write ≈ 11-15 us @ 23.3 TB/s. Compute is negligible.
// ---------------------------------------------------------------------------

#define B_  8
#define L_  4096
#define C_  1024
#define H_  512
#define LCHUNK 128
#define NLCH (L_ / LCHUNK)   // 32

typedef float v2f __attribute__((ext_vector_type(2)));
typedef float v4f __attribute__((ext_vector_type(4)));
typedef float v8f __attribute__((ext_vector_type(8)));

__device__ __forceinline__ float wave_sum32(float v) {
    #pragma unroll
    for (int off = 16; off > 0; off >>= 1) v += __shfl_xor(v, off, 32);
    return v;
}
__device__ __forceinline__ float wave_max32(float v) {
    #pragma unroll
    for (int off = 16; off > 0; off >>= 1) v = fmaxf(v, __shfl_xor(v, off, 32));
    return v;
}

// K1: t1[b*L+l] = dot(x[b,l,:], w1) + b1.  One wave32 per row, float4 loads.
__global__ void k1_rowdot(const float* __restrict__ x, const float* __restrict__ w1,
                          const float* __restrict__ b1, float* __restrict__ t1) {
    __shared__ float w1s[C_];
    const int tid = threadIdx.x;
    for (int i = tid; i < C_ / 4; i += blockDim.x)
        ((float4*)w1s)[i] = ((const float4*)w1)[i];
    __syncthreads();
    const int wave = tid >> 5, lane = tid & 31;
    const long long row = (long long)blockIdx.x * 8 + wave;   // 8 waves/block
    const float* xr = x + row * C_;
    float acc = 0.f;
    #pragma unroll
    for (int j = 0; j < C_; j += 128) {               // 128 floats per wave-step
        float4 xv = ((const float4*)(xr + j))[lane];
        float4 wv = ((const float4*)(w1s + j))[lane];
        acc += xv.x * wv.x + xv.y * wv.y + xv.z * wv.z + xv.w * wv.w;
    }
    acc = wave_sum32(acc);
    if (lane == 0) t1[row] = acc + b1[0];
}

// K2: softmax over L per batch. One block (256 thr) per batch.
__global__ void k2_softmax(const float* __restrict__ t1, float* __restrict__ a) {
    __shared__ float red[8];
    const int b = blockIdx.x, tid = threadIdx.x;
    const float* t = t1 + (long long)b * L_;
    float* ab = a + (long long)b * L_;
    float m = -3.402823e38f;
    for (int i = tid; i < L_; i += 256) m = fmaxf(m, t[i]);
    m = wave_max32(m);
    if ((tid & 31) == 0) red[tid >> 5] = m;
    __syncthreads();
    float bm = red[0];
    #pragma unroll
    for (int w = 1; w < 8; ++w) bm = fmaxf(bm, red[w]);
    __syncthreads();
    float s = 0.f;
    for (int i = tid; i < L_; i += 256) s += __expf(t[i] - bm);
    s = wave_sum32(s);
    if ((tid & 31) == 0) red[tid >> 5] = s;
    __syncthreads();
    float Z = 0.f;
    #pragma unroll
    for (int w = 0; w < 8; ++w) Z += red[w];
    const float inv = 1.f / Z;
    for (int i = tid; i < L_; i += 256) ab[i] = __expf(t[i] - bm) * inv;
}

// K3: partial weighted sums: part[lch][b][c] = sum_{l in chunk} a[b,l]*x[b,l,c]
// grid = (NLCH, B), 256 threads; each thread owns 4 channels (float4),
// block covers the full 1024-channel row -> 4 KB coalesced b128 reads per l.
__global__ void k3_wsum_part(const float* __restrict__ x, const float* __restrict__ a,
                             float* __restrict__ part) {
    __shared__ float as[LCHUNK];
    const int tid = threadIdx.x;
    const int lch = blockIdx.x, b = blockIdx.y;
    const int l0 = lch * LCHUNK;
    if (tid < LCHUNK) as[tid] = a[b * L_ + l0 + tid];
    __syncthreads();
    const float4* xb = (const float4*)(x + ((long long)b * L_ + l0) * C_) + tid;
    float4 acc = make_float4(0.f, 0.f, 0.f, 0.f);
    #pragma unroll 4
    for (int l = 0; l < LCHUNK; ++l) {
        float4 xv = xb[l * (C_ / 4)];
        const float w = as[l];
        acc.x += w * xv.x; acc.y += w * xv.y;
        acc.z += w * xv.z; acc.w += w * xv.w;
    }
    ((float4*)(part + ((long long)lch * B_ + b) * C_))[tid] = acc;
}

// K4: reduce partials over NLCH -> y[B*C]
__global__ void k4_reduce(const float* __restrict__ part, float* __restrict__ y) {
    const int i = blockIdx.x * blockDim.x + threadIdx.x;
    const int BC = B_ * C_;
    float s = 0.f;
    #pragma unroll
    for (int k = 0; k < NLCH; ++k) s += part[k * BC + i];
    y[i] = s;
}

// K5: out[8,N] = act(A[8,K] @ W[K,N] + bias), fp32 WMMA 16x16x4.
// One wave per 16-column tile; A "padded" to M=16 by duplicating rows 0-7
// into rows 8-15 (D rows 8-15 depend only on A rows 8-15 and are discarded).
// A frag (v2f): vgpr0 = K k0 (lanes 0-15) / k0+2 (lanes 16-31); vgpr1 = +1.
// B frag (v2f): row K striped across lanes (N = lane&15), same K split as A.
__global__ void k5_gemm8_wmma(const float* __restrict__ A, const float* __restrict__ W,
                              const float* __restrict__ bias, float* __restrict__ out,
                              int K, int N, int act) {
    const int lane = threadIdx.x & 31;
    const int wave = threadIdx.x >> 5;
    const int n0 = (blockIdx.x * (blockDim.x >> 5) + wave) * 16;
    const int half = lane >> 4;             // 0: lanes 0-15, 1: lanes 16-31
    const int l16 = lane & 15;
    const int m7 = l16 & 7;                 // duplicate A rows for M>=8
    const float* ap = A + m7 * K + half * 2;            // contiguous pair -> b64
    const float* bp = W + (half * 2) * N + n0 + l16;
    const int bstep = 4 * N;
    v8f c = {};
    for (int k0 = 0; k0 < K; k0 += 4) {
        v2f af = *(const v2f*)ap;           // A[m7][k0+2h], A[m7][k0+2h+1]
        v2f bf;
        bf.x = bp[0];                       // W[k0+2h    ][n0+l16]
        bf.y = bp[N];                       // W[k0+2h + 1][n0+l16]
        c = __builtin_amdgcn_wmma_f32_16x16x4_f32(false, af, false, bf,
                                                  (short)0, c, false, false);
        ap += 4;
        bp += bstep;
    }
    // C/D layout: vgpr i, lanes 0-15 hold row M=i (rows 0..7 are the real ones)
    if (half == 0) {
        const int n = n0 + l16;
        const float bv = bias[n];
        #pragma unroll
        for (int i = 0; i < 8; ++i) {
            float v = c[i] + bv;
            if (act) v = 1.f / (1.f + __expf(-v));
            out[i * N + n] = v;
        }
    }
}

// K6: sigma[b] = sigmoid(dot(v1[b,:], v2[b,:])) — one wave per batch.
__global__ void k6_dot_sigmoid(const float* __restrict__ v1, const float* __restrict__ v2,
                               float* __restrict__ sigma) {
    const int b = threadIdx.x >> 5, lane = threadIdx.x & 31;
    const float* p = v1 + b * H_;
    const float* q = v2 + b * H_;
    float s = 0.f;
    for (int h = lane; h < H_; h += 32) s += p[h] * q[h];
    s = wave_sum32(s);
    if (lane == 0) sigma[b] = 1.f / (1.f + __expf(-s));
}

// K7: out[b, :, :] = sigma[b] — nontemporal 128-bit broadcast store (128 MB).
__global__ void k7_broadcast(const float* __restrict__ sigma, float* __restrict__ out) {
    const int b = blockIdx.y;
    const float v = sigma[b];
    const long long i = (long long)blockIdx.x * blockDim.x + threadIdx.x;
    v4f* o = (v4f*)(out + (long long)b * L_ * C_) + i;
    v4f val = {v, v, v, v};
    __builtin_nontemporal_store(val, o);
}

extern "C" void kernel_launch(void* const* d_in, const int* in_sizes, int n_in,
                              void* d_out, int out_size, void* d_ws, size_t ws_size,
                              hipStream_t stream) {
    const float* x  = (const float*)d_in[0];
    const float* w1 = (const float*)d_in[1];
    const float* b1 = (const float*)d_in[2];
    const float* w2 = (const float*)d_in[3];
    const float* b2 = (const float*)d_in[4];
    const float* w3 = (const float*)d_in[5];
    const float* b3 = (const float*)d_in[6];
    const float* w4 = (const float*)d_in[7];
    const float* b4 = (const float*)d_in[8];
    const float* w5 = (const float*)d_in[9];
    const float* b5 = (const float*)d_in[10];
    float* out = (float*)d_out;

    // workspace layout (floats)
    float* ws = (float*)d_ws;
    float* t1    = ws;                                   // B*L      = 32768
    float* a     = t1 + (long long)B_ * L_;              // 32768
    float* part  = a  + (long long)B_ * L_;              // NLCH*B*C = 262144
    float* y     = part + (long long)NLCH * B_ * C_;     // B*C = 8192
    float* z     = y  + (long long)B_ * C_;              // B*H = 4096
    float* u     = z  + (long long)B_ * H_;              // B*C = 8192
    float* v1    = u  + (long long)B_ * C_;              // B*H = 4096
    float* v2    = v1 + (long long)B_ * H_;              // B*H = 4096
    float* sigma = v2 + (long long)B_ * H_;              // B   = 8

    // 1) t1 = x@w1 + b1            (streams 128 MB of x, fills L2)
    k1_rowdot<<<(B_ * L_) / 8, 256, 0, stream>>>(x, w1, b1, t1);
    // 2) a = softmax_L(t1)
    k2_softmax<<<B_, 256, 0, stream>>>(t1, a);
    // 3) y = sum_l a[l]*x[l,:]     (second 128 MB pass; should hit L2)
    k3_wsum_part<<<dim3(NLCH, B_), 256, 0, stream>>>(x, a, part);
    k4_reduce<<<(B_ * C_) / 256, 256, 0, stream>>>(part, y);
    // 4) tiny fp32 WMMA GEMM chain (M=8 padded to 16 by row duplication)
    k5_gemm8_wmma<<<(H_ / 16) / 8, 256, 0, stream>>>(y, w2, b2, z, C_, H_, 0); // z=y@w2+b2
    k5_gemm8_wmma<<<(C_ / 16) / 8, 256, 0, stream>>>(z, w3, b3, u, H_, C_, 1); // u=sig(z@w3+b3)
    k5_gemm8_wmma<<<(H_ / 16) / 8, 256, 0, stream>>>(u, w4, b4, v1, C_, H_, 0);
    k5_gemm8_wmma<<<(H_ / 16) / 8, 256, 0, stream>>>(u, w5, b5, v2, C_, H_, 0);
    // 5) sigma[b] = sigmoid(<v1,v2>)
    k6_dot_sigmoid<<<1, 256, 0, stream>>>(v1, v2, sigma);
    // 6) broadcast 128 MB nontemporal store
    k7_broadcast<<<dim3((L_ * C_ / 4) / 256, B_), 256, 0, stream>>>(sigma, out);
}